// Model_37701222924977
// MI455X (gfx1250) — compile-verified
//
#include <hip/hip_runtime.h>
#include <hip/hip_bf16.h>

typedef __attribute__((ext_vector_type(2))) float v2f;
typedef __attribute__((ext_vector_type(8))) float v8f;

#define SCORE_TH 0.2f
#define SIM_TH   0.9f

__device__ __forceinline__ unsigned long long packKey(float v, unsigned idx) {
  // v >= 0 after threshold -> float bits are order-preserving as u32.
  // low word = ~idx so max() prefers the SMALLEST index on value ties (jnp.argmax).
  return (((unsigned long long)__float_as_uint(v)) << 32) |
         (unsigned long long)(~idx);
}

__device__ __forceinline__ unsigned long long shflXor64(unsigned long long v, int mask) {
  unsigned lo = (unsigned)(v & 0xFFFFFFFFull);
  unsigned hi = (unsigned)(v >> 32);
  lo = __shfl_xor(lo, mask, 32);
  hi = __shfl_xor(hi, mask, 32);
  return (((unsigned long long)hi) << 32) | (unsigned long long)lo;
}

// ---------------- kernel 0: init argmax key slots ----------------
__global__ void init_keys(unsigned long long* __restrict__ g_row,
                          unsigned long long* __restrict__ g_col,
                          int M, int N) {
  int i = blockIdx.x * blockDim.x + threadIdx.x;
  if (i < M) g_row[i] = 0ull;
  if (i < N) g_col[i] = 0ull;
}

// ---------------- kernel 1: stable compaction (valid-first, stable) ----------------
__global__ __launch_bounds__(1024)
void compact_kernel(const float* __restrict__ scores,       // [N,1]
                    const float* __restrict__ keypoints,    // [N,2]
                    const float* __restrict__ descriptors,  // [N,64]
                    float* __restrict__ desc_c,             // [N,64]
                    float* __restrict__ kp_c,               // [N,2]
                    int N) {
  __shared__ int sums[1024];
  const int t = threadIdx.x;
  const int per = N / 1024;
  const int base = t * per;

  int cnt = 0;
  for (int i = 0; i < per; ++i) cnt += (scores[base + i] > SCORE_TH) ? 1 : 0;
  sums[t] = cnt;
  __syncthreads();
  // Hillis-Steele inclusive scan over 1024 per-thread counts
  for (int off = 1; off < 1024; off <<= 1) {
    int x = (t >= off) ? sums[t - off] : 0;
    __syncthreads();
    sums[t] += x;
    __syncthreads();
  }
  const int totalValid = sums[1023];
  int run = sums[t] - cnt;  // exclusive prefix of valid count at element `base`

  for (int i = 0; i < per; ++i) {
    const int idx = base + i;
    const bool valid = scores[idx] > SCORE_TH;
    // stable: valid rows compact to front in order; invalid rows follow in order
    const int pos = valid ? run : (totalValid + (idx - run));
    kp_c[2 * pos]     = keypoints[2 * idx];
    kp_c[2 * pos + 1] = keypoints[2 * idx + 1];
    const float4* s = (const float4*)(descriptors + (size_t)idx * 64);
    float4* d = (float4*)(desc_c + (size_t)pos * 64);
    if (valid) {
      #pragma unroll
      for (int k = 0; k < 16; ++k) d[k] = s[k];
      run++;
    } else {
      float4 z = make_float4(0.f, 0.f, 0.f, 0.f);
      #pragma unroll
      for (int k = 0; k < 16; ++k) d[k] = z;
    }
  }
}

// ---------------- kernel 2: fused WMMA GEMM + threshold + dual argmax ----------------
// sim[m,n] = sum_k prev[m,k] * desc_c[n,k]; block tile 128x128, 8 waves,
// wave w owns rows [m0+16w, m0+16w+16) x all 128 cols = 8 f32 16x16 accumulators.
// All fragment loads use ONE base address per matrix with compile-time immediate
// offsets (t*1024 + k0 floats), and are double-buffered so loads for step k+1
// overlap the 8 WMMAs of step k.
#define BM 128
#define BN 128

__global__ __launch_bounds__(256)
void sim_argmax_kernel(const float* __restrict__ A,   // prev_descriptors [M,64]
                       const float* __restrict__ B,   // desc_c           [N,64]
                       unsigned long long* __restrict__ g_row,
                       unsigned long long* __restrict__ g_col,
                       int M, int N) {
  __shared__ unsigned long long lds_col[BN];

  const int tid  = threadIdx.x;
  const int wave = tid >> 5;
  const int lane = tid & 31;
  const int half = lane >> 4;   // K-pair select for A/B fragments
  const int lrow = lane & 15;   // M index (A frag) / N index (B frag)

  const int m0 = blockIdx.y * BM;
  const int n0 = blockIdx.x * BN;
  const int mw = m0 + wave * 16;

  if (tid < BN) lds_col[tid] = 0ull;
  __syncthreads();

  v8f acc[8];
  #pragma unroll
  for (int t = 0; t < 8; ++t)
    #pragma unroll
    for (int j = 0; j < 8; ++j) acc[t][j] = 0.0f;

  // A fragment (16x4 f32): lane (half,lrow) holds A[mw+lrow][k0+2*half .. +1]
  // B fragment (4x16 f32): lane (half,lrow) holds desc_c[n0+16t+lrow][k0+2*half .. +1]
  // Loop-invariant bases; per-step/per-tile offsets are immediates after unroll.
  const float* abase = A + (size_t)(mw + lrow) * 64 + 2 * half;
  const float* bbase = B + (size_t)(n0 + lrow) * 64 + 2 * half;

  float2 a_cur = *(const float2*)(abase);
  float2 b_cur[8];
  #pragma unroll
  for (int t = 0; t < 8; ++t) b_cur[t] = *(const float2*)(bbase + t * 1024);

  #pragma unroll
  for (int k0 = 0; k0 < 64; k0 += 4) {
    float2 a_nxt;
    float2 b_nxt[8];
    if (k0 < 60) {
      a_nxt = *(const float2*)(abase + (k0 + 4));
      #pragma unroll
      for (int t = 0; t < 8; ++t)
        b_nxt[t] = *(const float2*)(bbase + t * 1024 + (k0 + 4));
    }
    v2f a; a.x = a_cur.x; a.y = a_cur.y;
    #pragma unroll
    for (int t = 0; t < 8; ++t) {
      v2f b; b.x = b_cur[t].x; b.y = b_cur[t].y;
      acc[t] = __builtin_amdgcn_wmma_f32_16x16x4_f32(
          /*neg_a=*/false, a, /*neg_b=*/false, b,
          /*c_mod=*/(short)0, acc[t], /*reuse_a=*/false, /*reuse_b=*/false);
    }
    if (k0 < 60) {
      a_cur = a_nxt;
      #pragma unroll
      for (int t = 0; t < 8; ++t) b_cur[t] = b_nxt[t];
    }
  }

  // threshold: sim < 0.9 -> 0
  #pragma unroll
  for (int t = 0; t < 8; ++t)
    #pragma unroll
    for (int j = 0; j < 8; ++j)
      if (acc[t][j] < SIM_TH) acc[t][j] = 0.0f;

  // C/D layout: vgpr j, lanes 0-15: row=j,   col=lrow
  //                     lanes 16-31: row=j+8, col=lrow
  // ---- row argmax (over this wave's 128 columns) ----
  #pragma unroll
  for (int j = 0; j < 8; ++j) {
    unsigned long long key = 0ull;
    #pragma unroll
    for (int t = 0; t < 8; ++t) {
      unsigned long long k = packKey(acc[t][j], (unsigned)(n0 + 16 * t + lrow));
      if (k > key) key = k;
    }
    // reduce across the 16 lanes of each half (masks 1..8 stay within half)
    #pragma unroll
    for (int msk = 1; msk <= 8; msk <<= 1) {
      unsigned long long o = shflXor64(key, msk);
      if (o > key) key = o;
    }
    if (lrow == 0) atomicMax(&g_row[mw + j + 8 * half], key);
  }

  // ---- col argmax (over this wave's 16 rows), merged across waves in LDS ----
  #pragma unroll
  for (int t = 0; t < 8; ++t) {
    unsigned long long key = 0ull;
    #pragma unroll
    for (int j = 0; j < 8; ++j) {
      unsigned long long k = packKey(acc[t][j], (unsigned)(mw + j + 8 * half));
      if (k > key) key = k;
    }
    // lanes L and L+16 hold the same column (rows j vs j+8): merge halves
    unsigned long long o = shflXor64(key, 16);
    if (o > key) key = o;
    if (half == 0) atomicMax(&lds_col[16 * t + lrow], key);
  }
  __syncthreads();
  if (tid < BN) atomicMax(&g_col[n0 + tid], lds_col[tid]);
}

// ---------------- kernel 3: mutual-NN decode + padded outputs ----------------
__global__ void finalize_kernel(const unsigned long long* __restrict__ g_row,
                                const unsigned long long* __restrict__ g_col,
                                const float* __restrict__ prev_kp,  // [M,2]
                                const float* __restrict__ kp_c,     // [N,2]
                                float* __restrict__ out, int M) {
  int m = blockIdx.x * blockDim.x + threadIdx.x;
  if (m >= M) return;
  unsigned nn12  = ~(unsigned)(g_row[m] & 0xFFFFFFFFull);
  unsigned nn21v = ~(unsigned)(g_col[nn12] & 0xFFFFFFFFull);
  const bool match = (nn21v == (unsigned)m);

  float* out1 = out;           // matched_keypoints1 [M,2]
  float* out2 = out + 2 * M;   // matched_keypoints2 [M,2]
  float* outm = out + 4 * M;   // match [M]

  out1[2 * m]     = match ? prev_kp[2 * m]     : 0.0f;
  out1[2 * m + 1] = match ? prev_kp[2 * m + 1] : 0.0f;
  out2[2 * m]     = match ? kp_c[2 * nn12]     : 0.0f;
  out2[2 * m + 1] = match ? kp_c[2 * nn12 + 1] : 0.0f;
  outm[m] = match ? 1.0f : 0.0f;
}

extern "C" void kernel_launch(void* const* d_in, const int* in_sizes, int n_in,
                              void* d_out, int out_size, void* d_ws, size_t ws_size,
                              hipStream_t stream) {
  const float* keypoints        = (const float*)d_in[0];  // [N,2]
  const float* descriptors      = (const float*)d_in[1];  // [N,64]
  const float* scores           = (const float*)d_in[2];  // [N,1]
  const float* prev_keypoints   = (const float*)d_in[3];  // [M,2]
  const float* prev_descriptors = (const float*)d_in[4];  // [M,64]
  const int N = in_sizes[0] / 2;
  const int M = in_sizes[3] / 2;

  // workspace layout (all 8B-aligned)
  char* ws = (char*)d_ws;
  float* desc_c = (float*)ws;                                   // N*64 f32 (4 MB)
  float* kp_c   = (float*)(ws + (size_t)N * 64 * 4);            // N*2  f32
  unsigned long long* g_row =
      (unsigned long long*)(ws + (size_t)N * 66 * 4);           // M u64
  unsigned long long* g_col = g_row + M;                        // N u64

  const int initN = (M > N) ? M : N;
  init_keys<<<(initN + 255) / 256, 256, 0, stream>>>(g_row, g_col, M, N);
  compact_kernel<<<1, 1024, 0, stream>>>(scores, keypoints, descriptors,
                                         desc_c, kp_c, N);
  dim3 grid(N / BN, M / BM);
  sim_argmax_kernel<<<grid, 256, 0, stream>>>(prev_descriptors, desc_c,
                                              g_row, g_col, M, N);
  finalize_kernel<<<(M + 255) / 256, 256, 0, stream>>>(g_row, g_col,
                                                       prev_keypoints, kp_c,
                                                       (float*)d_out, M);
}